// Router_4964982194280
// MI455X (gfx1250) — compile-verified
//
#include <hip/hip_runtime.h>
#include <stdint.h>

#define HIDDEN      4096
#define NUM_EXPERTS 64
#define TOKENS      16384
#define MB          32             // tokens per block
#define KC          128            // K-chunk per TDM stage
#define NCHUNK      (HIDDEN / KC)  // 32
#define LDS_STRIDE  132            // 128 dwords + 4 dword TDM pad per row

typedef __attribute__((ext_vector_type(2))) float v2f;
typedef __attribute__((ext_vector_type(4))) float v4f;
typedef __attribute__((ext_vector_type(8))) float v8f;
typedef __attribute__((ext_vector_type(4))) unsigned int v4u;
typedef __attribute__((ext_vector_type(4))) int v4i;
typedef __attribute__((ext_vector_type(8))) int v8i;

// Build a 2-D tensor-DMA descriptor (D#) per CDNA5 ISA §8.3/§8.4 and issue
// TENSOR_LOAD_TO_LDS. Pads LDS destination rows: +4 dwords every 128 dwords
// (pad_interval code 6 = 128 dw, pad_amount code 3 = 4 dw) -> row stride 132.
static __device__ __forceinline__ void tdm_load_2d(unsigned lds_off,
                                                   const void* gptr,
                                                   unsigned tile_d0,
                                                   unsigned tile_d1,
                                                   unsigned tensor_d0,
                                                   unsigned tensor_d1,
                                                   unsigned stride0)
{
    const uint64_t ga = (uint64_t)gptr;
    v4u g0;
    v8i g1;
    v4i g2 = {0, 0, 0, 0};                 // 2-D tile: dims 2..4 unused
    v4i g3 = {0, 0, 0, 0};
    v8i g4 = {0, 0, 0, 0, 0, 0, 0, 0};     // extra group in 6-arg builtin form

    g0[0] = 1u;                                                // count=1, user desc
    g0[1] = lds_off;                                           // lds_addr (bytes)
    g0[2] = (unsigned)ga;                                      // global_addr[31:0]
    g0[3] = (unsigned)((ga >> 32) & 0x1FFFFFFu) | (2u << 30);  // addr[56:32] | type=2

    const unsigned w0 = (2u << 16)    // data_size = 4 bytes
                      | (1u << 20)    // pad_enable
                      | (6u << 22)    // pad_interval = 128 dwords
                      | (3u << 25);   // pad_amount   = 4 dwords
    g1[0] = (int)w0;                                                        // mask=0
    g1[1] = (int)((tensor_d0 & 0xFFFFu) << 16);                             // dim0 lo
    g1[2] = (int)(((tensor_d0 >> 16) & 0xFFFFu) | ((tensor_d1 & 0xFFFFu) << 16));
    g1[3] = (int)(((tensor_d1 >> 16) & 0xFFFFu) | ((tile_d0 & 0xFFFFu) << 16));
    g1[4] = (int)(tile_d1 & 0xFFFFu);                                       // tile_dim1
    g1[5] = (int)stride0;                                                   // stride0 lo
    g1[6] = 0;                                                              // stride0 hi
    g1[7] = 0;

    __builtin_amdgcn_tensor_load_to_lds(g0, g1, g2, g3, g4, 0);
}

// Block = 32 tokens x 64 experts, 4 wave32s. Wave w owns M-tile (w>>1) and the
// 32-expert pair (w&1), holding two 16x16 f32 accumulators so each A fragment
// feeds two WMMAs: 3 ds_load_b128 per 4 v_wmma_f32_16x16x4_f32.
__global__ __launch_bounds__(128, 2)
void router_wmma_tdm(const float* __restrict__ x,
                     const float* __restrict__ wgt,
                     float* __restrict__ out_w,
                     int*   __restrict__ out_i)
{
    __shared__ float xtile[2][MB][LDS_STRIDE];            // 2 x 16.5 KB
    __shared__ float wtile[2][NUM_EXPERTS][LDS_STRIDE];   // 2 x 33 KB
    __shared__ float logits[MB][NUM_EXPERTS + 1];

    const int wave  = threadIdx.x >> 5;
    const int lane  = threadIdx.x & 31;
    const int half  = lane >> 4;       // K sub-slot selector
    const int sub   = lane & 15;       // row (token / expert) within 16-tile
    const int mtile = wave >> 1;       // which 16-token half of the block
    const int npair = wave & 1;        // which 32-expert half
    const int row0  = blockIdx.x * MB;

    if (wave == 0) {
        tdm_load_2d((unsigned)(size_t)&xtile[0][0][0],
                    x + (size_t)row0 * HIDDEN, KC, MB, HIDDEN, MB, HIDDEN);
        tdm_load_2d((unsigned)(size_t)&wtile[0][0][0],
                    wgt, KC, NUM_EXPERTS, HIDDEN, NUM_EXPERTS, HIDDEN);
    }

    v8f acc0 = {};
    v8f acc1 = {};
    for (int c = 0; c < NCHUNK; ++c) {
        const int buf = c & 1;
        if (wave == 0)
            __builtin_amdgcn_s_wait_tensorcnt(0);   // chunk c resident in LDS
        __syncthreads();                            // release all waves; buf^1 free
        if (wave == 0 && (c + 1) < NCHUNK) {
            const int k = (c + 1) * KC;
            tdm_load_2d((unsigned)(size_t)&xtile[buf ^ 1][0][0],
                        x + (size_t)row0 * HIDDEN + k, KC, MB, HIDDEN, MB, HIDDEN);
            tdm_load_2d((unsigned)(size_t)&wtile[buf ^ 1][0][0],
                        wgt + k, KC, NUM_EXPERTS, HIDDEN, NUM_EXPERTS, HIDDEN);
        }

        // K-slot permutation: half h takes K = {kk+4h, kk+4h+1} (slot .xy) and
        // {kk+4h+2, kk+4h+3} (slot .zw); identical for A and B, so the K-sum is
        // unchanged. One A b128 + two B b128 LDS reads feed four WMMAs.
        const float* ar  = &xtile[buf][(mtile << 4) + sub][4 * half];
        const float* br0 = &wtile[buf][(npair << 5) + sub][4 * half];
        const float* br1 = &wtile[buf][(npair << 5) + 16 + sub][4 * half];
#pragma unroll
        for (int kk = 0; kk < KC; kk += 8) {
            const v4f a4  = *(const v4f*)(ar  + kk);
            const v4f b40 = *(const v4f*)(br0 + kk);
            const v4f b41 = *(const v4f*)(br1 + kk);
            acc0 = __builtin_amdgcn_wmma_f32_16x16x4_f32(
                false, a4.xy, false, b40.xy, (short)0, acc0, false, false);
            acc0 = __builtin_amdgcn_wmma_f32_16x16x4_f32(
                false, a4.zw, false, b40.zw, (short)0, acc0, false, false);
            acc1 = __builtin_amdgcn_wmma_f32_16x16x4_f32(
                false, a4.xy, false, b41.xy, (short)0, acc1, false, false);
            acc1 = __builtin_amdgcn_wmma_f32_16x16x4_f32(
                false, a4.zw, false, b41.zw, (short)0, acc1, false, false);
        }
    }

    // C/D layout: acc*[r] holds M = 16*mtile + r + 8*half, N = sub (+16 for acc1)
#pragma unroll
    for (int r = 0; r < 8; ++r) {
        logits[(mtile << 4) + r + 8 * half][(npair << 5) + sub]      = acc0[r];
        logits[(mtile << 4) + r + 8 * half][(npair << 5) + 16 + sub] = acc1[r];
    }

    __syncthreads();

    if (threadIdx.x < MB) {
        const int m = threadIdx.x;
        float v1 = -__builtin_inff(), v2 = -__builtin_inff();
        int   i1 = 0, i2 = 0;
        for (int e = 0; e < NUM_EXPERTS; ++e) {
            float v = logits[m][e];
            if (v > v1)      { v2 = v1; i2 = i1; v1 = v; i1 = e; }
            else if (v > v2) { v2 = v;  i2 = e; }
        }
        const float w1 = 1.0f / (1.0f + __expf(v2 - v1));
        const float w2 = 1.0f - w1;
        const int t = row0 + m;
        out_w[2 * t + 0] = w1;
        out_w[2 * t + 1] = w2;
        out_i[2 * t + 0] = i1;
        out_i[2 * t + 1] = i2;
    }
}

extern "C" void kernel_launch(void* const* d_in, const int* in_sizes, int n_in,
                              void* d_out, int out_size, void* d_ws, size_t ws_size,
                              hipStream_t stream)
{
    (void)in_sizes; (void)n_in; (void)out_size; (void)d_ws; (void)ws_size;
    const float* x   = (const float*)d_in[0];
    const float* wgt = (const float*)d_in[1];
    float* out_w = (float*)d_out;
    int*   out_i = (int*)d_out + (size_t)TOKENS * 2;

    router_wmma_tdm<<<TOKENS / MB, 128, 0, stream>>>(x, wgt, out_w, out_i);
}